// GATBlock_54417235640492
// MI455X (gfx1250) — compile-verified
//
#include <hip/hip_runtime.h>
#include <math.h>

typedef __attribute__((ext_vector_type(2))) float v2f;
typedef __attribute__((ext_vector_type(8))) float v8f;

#define HC 128
#define HEADS 4
#define CH 32
#define EDIM 13
#define NEG_SLOPE 0.2f
#define LN_EPS 1e-5f

// ---------- helpers ----------
__device__ inline void atomicMaxF(float* addr, float val) {
    unsigned int* ia = (unsigned int*)addr;
    unsigned int old = *ia;
    while (__uint_as_float(old) < val) {
        unsigned int assumed = old;
        old = atomicCAS(ia, assumed, __float_as_uint(val));
        if (old == assumed) break;
    }
}

// ---------- kernel 0: fill ----------
__global__ void k_fill(float* p, float v, int n) {
    int i = blockIdx.x * blockDim.x + threadIdx.x;
    if (i < n) p[i] = v;
}

// ---------- kernel 1: self-loop stats (count + edge_attr sum per dst) ----------
__global__ void k_loop_stats(const long long* __restrict__ ei, const float* __restrict__ ea,
                             float* __restrict__ cnt, float* __restrict__ easum, int E) {
    int e = blockIdx.x * blockDim.x + threadIdx.x;
    if (e >= E) return;
    int dst = (int)ei[(long long)E + e];
    atomicAdd(&cnt[dst], 1.0f);
    float* s = easum + (long long)dst * EDIM;
    const float* a = ea + (long long)e * EDIM;
#pragma unroll
    for (int d = 0; d < EDIM; ++d) atomicAdd(&s[d], a[d]);
}

// ---------- kernel 2: divide to get mean self-loop attr ----------
__global__ void k_loop_mean(float* __restrict__ easum, const float* __restrict__ cnt, int N) {
    int i = blockIdx.x * blockDim.x + threadIdx.x;
    if (i >= N * EDIM) return;
    int n = i / EDIM;
    easum[i] /= fmaxf(cnt[n], 1.0f);
}

// ---------- kernel 3: h = x @ W via V_WMMA_F32_16X16X4_F32 ----------
// Block: 256 threads = 8 waves; block handles 16 rows of x, each wave a 16x16 col tile.
__global__ void k_gemm_h(const float* __restrict__ x, const float* __restrict__ W,
                         float* __restrict__ h, int N) {
    __shared__ float ldsx[16 * HC];  // 8 KB x-tile
    int row0 = blockIdx.x * 16;
    int t = threadIdx.x;
    // cooperative float4 load of the 16x128 tile (512 float4, 2 per thread)
#pragma unroll
    for (int i = 0; i < 2; ++i) {
        int f4 = t + i * 256;
        int r = f4 >> 5, c4 = f4 & 31;
        int gr = row0 + r;
        if (gr >= N) gr = N - 1;
        float4 v = reinterpret_cast<const float4*>(x + (long long)gr * HC)[c4];
        reinterpret_cast<float4*>(ldsx)[f4] = v;
    }
    __syncthreads();

    int wave = t >> 5;
    int lane = t & 31;
    int col0 = wave * 16;
    int m    = lane & 15;   // A row / lane group
    int khi  = lane >> 4;   // 0: K={0,1}, 1: K={2,3} per ISA 16x4 f32 A layout
    int nn   = lane & 15;   // B/D column within tile

    v8f acc = {};
    for (int k0 = 0; k0 < HC; k0 += 4) {
        v2f a, b;
        a.x = ldsx[m * HC + k0 + 2 * khi];
        a.y = ldsx[m * HC + k0 + 2 * khi + 1];
        b.x = W[(long long)(k0 + 2 * khi) * HC + col0 + nn];
        b.y = W[(long long)(k0 + 2 * khi + 1) * HC + col0 + nn];
        acc = __builtin_amdgcn_wmma_f32_16x16x4_f32(
            /*neg_a=*/false, a, /*neg_b=*/false, b,
            /*c_mod=*/(short)0, acc, /*reuse_a=*/false, /*reuse_b=*/false);
    }
    // D layout: VGPR v -> row (v + 8*khi), col nn
#pragma unroll
    for (int v = 0; v < 8; ++v) {
        int gr = row0 + v + 8 * khi;
        if (gr < N) h[(long long)gr * HC + col0 + nn] = acc[v];
    }
}

// ---------- kernel 4: per-(node,head) attention dots ----------
__global__ void k_att_nodes(const float* __restrict__ h, const float* __restrict__ att_src,
                            const float* __restrict__ att_dst, float* __restrict__ a_src,
                            float* __restrict__ a_dst, int N) {
    int i = blockIdx.x * blockDim.x + threadIdx.x;
    if (i >= N * HEADS) return;
    int n = i >> 2, hd = i & 3;
    const float* hp = h + (long long)n * HC + hd * CH;
    float s = 0.f, d = 0.f;
#pragma unroll
    for (int c = 0; c < CH; ++c) {
        float v = hp[c];
        s += v * att_src[hd * CH + c];
        d += v * att_dst[hd * CH + c];
    }
    a_src[i] = s;
    a_dst[i] = d;
}

// ---------- kernel 5: fold W_edge with att_edge -> v_edge[13][4] ----------
__global__ void k_vedge(const float* __restrict__ W_edge, const float* __restrict__ att_edge,
                        float* __restrict__ v_edge) {
    int t = threadIdx.x;
    if (t >= EDIM * HEADS) return;
    int d = t >> 2, hd = t & 3;
    float s = 0.f;
#pragma unroll
    for (int c = 0; c < CH; ++c) s += W_edge[d * HC + hd * CH + c] * att_edge[hd * CH + c];
    v_edge[t] = s;
}

// ---------- kernel 6: logits + LeakyReLU + segment max ----------
__global__ void k_alpha_max(const long long* __restrict__ ei, const float* __restrict__ ea,
                            const float* __restrict__ ealoop, const float* __restrict__ a_src,
                            const float* __restrict__ a_dst, const float* __restrict__ v_edge,
                            float* __restrict__ alpha, float* __restrict__ amax, int E, int N) {
    int e = blockIdx.x * blockDim.x + threadIdx.x;
    int T = E + N;
    if (e >= T) return;
    int src, dst;
    const float* ap;
    if (e < E) {
        src = (int)ei[e];
        dst = (int)ei[(long long)E + e];
        ap = ea + (long long)e * EDIM;
    } else {
        src = dst = e - E;
        ap = ealoop + (long long)(e - E) * EDIM;
    }
    float ed[EDIM];
#pragma unroll
    for (int d = 0; d < EDIM; ++d) ed[d] = ap[d];
#pragma unroll
    for (int hd = 0; hd < HEADS; ++hd) {
        float al = a_src[src * 4 + hd] + a_dst[dst * 4 + hd];
#pragma unroll
        for (int d = 0; d < EDIM; ++d) al += ed[d] * v_edge[d * 4 + hd];
        al = al > 0.f ? al : NEG_SLOPE * al;
        alpha[(long long)e * 4 + hd] = al;
        atomicMaxF(&amax[dst * 4 + hd], al);
    }
}

// ---------- kernel 7: exp + segment sum ----------
__global__ void k_exp_sum(const long long* __restrict__ ei, float* __restrict__ alpha,
                          const float* __restrict__ amax, float* __restrict__ denom, int E, int N) {
    int e = blockIdx.x * blockDim.x + threadIdx.x;
    int T = E + N;
    if (e >= T) return;
    int dst = (e < E) ? (int)ei[(long long)E + e] : (e - E);
#pragma unroll
    for (int hd = 0; hd < HEADS; ++hd) {
        float ex = __expf(alpha[(long long)e * 4 + hd] - amax[dst * 4 + hd]);
        alpha[(long long)e * 4 + hd] = ex;
        atomicAdd(&denom[dst * 4 + hd], ex);
    }
}

// ---------- kernel 8: weighted scatter-add aggregation (wave per edge) ----------
__global__ void k_aggregate(const long long* __restrict__ ei, const float* __restrict__ alpha,
                            const float* __restrict__ denom, const float* __restrict__ h,
                            float* __restrict__ out, int E, int N) {
    int t = blockIdx.x * blockDim.x + threadIdx.x;
    int e = t >> 5;
    int lane = t & 31;
    int T = E + N;
    if (e >= T) return;
    int src, dst;
    if (e < E) {
        src = (int)ei[e];
        dst = (int)ei[(long long)E + e];
    } else {
        src = dst = e - E;
    }
    const float* hp = h + (long long)src * HC;
    __builtin_prefetch(hp + lane, 0, 0);  // global_prefetch_b8: warm L2/L0 for gather row
    float* op = out + (long long)dst * HC;
#pragma unroll
    for (int hd = 0; hd < HEADS; ++hd) {
        float w = alpha[(long long)e * 4 + hd] / (denom[dst * 4 + hd] + 1e-16f);
        atomicAdd(&op[hd * CH + lane], w * hp[hd * CH + lane]);
    }
}

// ---------- kernel 9: bias + residual + LayerNorm + ReLU (wave per node) ----------
__global__ void k_finalize(const float* __restrict__ x, const float* __restrict__ bias,
                           const float* __restrict__ gamma, const float* __restrict__ beta,
                           float* __restrict__ out, int N) {
    int wid = threadIdx.x >> 5;
    int lane = threadIdx.x & 31;
    int n = blockIdx.x * 8 + wid;
    if (n >= N) return;
    float v[4];
    float s = 0.f;
#pragma unroll
    for (int j = 0; j < 4; ++j) {
        int c = j * 32 + lane;
        v[j] = out[(long long)n * HC + c] + bias[c] + x[(long long)n * HC + c];
        s += v[j];
    }
#pragma unroll
    for (int off = 16; off > 0; off >>= 1) s += __shfl_xor(s, off, 32);
    float mu = s * (1.0f / HC);
    float q = 0.f;
#pragma unroll
    for (int j = 0; j < 4; ++j) {
        float d = v[j] - mu;
        q += d * d;
    }
#pragma unroll
    for (int off = 16; off > 0; off >>= 1) q += __shfl_xor(q, off, 32);
    float rstd = rsqrtf(q * (1.0f / HC) + LN_EPS);
#pragma unroll
    for (int j = 0; j < 4; ++j) {
        int c = j * 32 + lane;
        float y = (v[j] - mu) * rstd * gamma[c] + beta[c];
        out[(long long)n * HC + c] = fmaxf(y, 0.f);
    }
}

// ---------- launch ----------
extern "C" void kernel_launch(void* const* d_in, const int* in_sizes, int n_in,
                              void* d_out, int out_size, void* d_ws, size_t ws_size,
                              hipStream_t stream) {
    const float*     x        = (const float*)d_in[0];
    const long long* ei       = (const long long*)d_in[1];
    const float*     ea       = (const float*)d_in[2];
    const float*     W        = (const float*)d_in[3];
    const float*     att_src  = (const float*)d_in[4];
    const float*     att_dst  = (const float*)d_in[5];
    const float*     W_edge   = (const float*)d_in[6];
    const float*     att_edge = (const float*)d_in[7];
    const float*     bias     = (const float*)d_in[8];
    const float*     gamma    = (const float*)d_in[9];
    const float*     beta     = (const float*)d_in[10];

    int N = in_sizes[0] / HC;
    int E = in_sizes[1] / 2;
    int T = E + N;

    char* ws = (char*)d_ws;
    size_t off = 0;
    auto carve = [&](size_t bytes) -> char* {
        char* p = ws + off;
        off += (bytes + 255) & ~(size_t)255;
        return p;
    };
    float* cnt    = (float*)carve((size_t)N * 4);
    float* ealoop = (float*)carve((size_t)N * EDIM * 4);
    float* hbuf   = (float*)carve((size_t)N * HC * 4);
    float* asrc   = (float*)carve((size_t)N * HEADS * 4);
    float* adst   = (float*)carve((size_t)N * HEADS * 4);
    float* vedge  = (float*)carve((size_t)EDIM * HEADS * 4);
    float* alpha  = (float*)carve((size_t)T * HEADS * 4);
    float* amax   = (float*)carve((size_t)N * HEADS * 4);
    float* denom  = (float*)carve((size_t)N * HEADS * 4);
    float* outf   = (float*)d_out;

    hipMemsetAsync(cnt, 0, (size_t)N * 4, stream);
    hipMemsetAsync(ealoop, 0, (size_t)N * EDIM * 4, stream);
    hipMemsetAsync(denom, 0, (size_t)N * HEADS * 4, stream);
    hipMemsetAsync(d_out, 0, (size_t)N * HC * 4, stream);
    k_fill<<<(N * HEADS + 255) / 256, 256, 0, stream>>>(amax, -3.402823466e38f, N * HEADS);

    k_loop_stats<<<(E + 255) / 256, 256, 0, stream>>>(ei, ea, cnt, ealoop, E);
    k_loop_mean<<<(N * EDIM + 255) / 256, 256, 0, stream>>>(ealoop, cnt, N);

    k_gemm_h<<<(N + 15) / 16, 256, 0, stream>>>(x, W, hbuf, N);
    k_att_nodes<<<(N * HEADS + 255) / 256, 256, 0, stream>>>(hbuf, att_src, att_dst, asrc, adst, N);
    k_vedge<<<1, 64, 0, stream>>>(W_edge, att_edge, vedge);

    k_alpha_max<<<(T + 255) / 256, 256, 0, stream>>>(ei, ea, ealoop, asrc, adst, vedge, alpha, amax, E, N);
    k_exp_sum<<<(T + 255) / 256, 256, 0, stream>>>(ei, alpha, amax, denom, E, N);

    long long aggThreads = (long long)T * 32;
    k_aggregate<<<(unsigned)((aggThreads + 255) / 256), 256, 0, stream>>>(ei, alpha, denom, hbuf, outf, E, N);

    k_finalize<<<(N + 7) / 8, 256, 0, stream>>>(x, bias, gamma, beta, outf, N);
}